// HyperdimensionalAttention_51049981280861
// MI455X (gfx1250) — compile-verified
//
#include <hip/hip_runtime.h>

// ---------------------------------------------------------------------------
// HyperdimensionalAttention for MI455X (gfx1250, wave32, WMMA)
// B=1, S=512, HID=512, NH=8, HD=64, DHYP=10000
//
// Algebraic restructure: the DHYP=10000 hyperdimension only appears in
// self-contractions, so it collapses into 64x64 Gram matrices:
//   G_kq = k_base @ q_base^T     (Q @ G_qk == Q @ G_kq^T)
//   G_qq = q_base @ q_base^T,  G_kk = k_base @ k_base^T   (row norms)
//   Mt   = Wh @ v_base^T         (V @ (v_base @ Wh^T) == V @ Mt^T)
// ~90 GFLOP / ~0.5 GB of hyperspace work collapses to ~2 GFLOP over ~25 MB
// -> bandwidth/latency bound; keep exact f32 via V_WMMA_F32_16X16X4_F32.
//
// GEMM tiling: one wave computes a 16x64 output tile (4 f32 accumulators).
// The A fragment is loaded once per k-step and reused by 4 WMMAs -> 4 WMMAs
// per 5 x b64 loads instead of 1 per 2, and the 4 independent WMMAs give the
// scheduler work to overlap with the next iteration's loads.
// ---------------------------------------------------------------------------

typedef float v2f __attribute__((ext_vector_type(2)));
typedef float v8f __attribute__((ext_vector_type(8)));

#define SEQ   512
#define HIDN  512
#define NHEAD 8
#define HDIM  64
#define DHYP_ 10000

// Generic batched GEMM on V_WMMA_F32_16X16X4_F32, 16x64 tile per wave.
//  BT=true : D = A @ B^T   (A [M,K] lda, B [N,K] ldb, both row-major)
//  BT=false: D = A @ B     (B [K,N] ldb row-major)
// Requirements: M % 16 == 0, N % 64 == 0, Kin % (4*gridDim.y) == 0.
// bias (nullable) indexed by output column n in [0,N).
// Split-K via gridDim.y (atomicAdd path; D pre-zeroed, bias null).
// blockDim must be 32 (WMMA needs EXEC all-ones).
template <bool BT>
__global__ __launch_bounds__(32) void gemm_wmma(
    const float* __restrict__ A, int lda, int strideA,
    const float* __restrict__ B, int ldb, int strideB,
    float* __restrict__ D, int ldd, int strideD,
    const float* __restrict__ bias, int M, int N, int Kin)
{
    const int ntn = N >> 6;                   // 64-wide column tiles
    const int tm  = (int)blockIdx.x / ntn;
    const int tn  = (int)blockIdx.x % ntn;
    const int m0  = tm << 4;
    const int n0  = tn << 6;
    const int z   = blockIdx.z;
    A += z * strideA;
    B += z * strideB;
    D += z * strideD;

    const int lane = threadIdx.x & 31;
    const int hlf  = lane >> 4;   // 0 -> K pair {0,1}; 1 -> K pair {2,3}
    const int l16  = lane & 15;

    const int kslice = Kin / gridDim.y;       // multiple of 4 by construction
    const int k0     = blockIdx.y * kslice;
    const int kend   = k0 + kslice;

    // A fragment: lane l16 owns matrix row m0+l16; K pair selected by hlf.
    const float* aRow = A + (m0 + l16) * lda + (hlf << 1);

    v8f acc0 = {}, acc1 = {}, acc2 = {}, acc3 = {};

    if (BT) {
        const float* bR0 = B + (n0 + l16) * ldb + (hlf << 1);
        const float* bR1 = bR0 + 16 * ldb;
        const float* bR2 = bR0 + 32 * ldb;
        const float* bR3 = bR0 + 48 * ldb;
        #pragma unroll 2
        for (int k = k0; k < kend; k += 4) {
            v2f a  = *(const v2f*)(aRow + k);
            v2f b0 = *(const v2f*)(bR0 + k);
            v2f b1 = *(const v2f*)(bR1 + k);
            v2f b2 = *(const v2f*)(bR2 + k);
            v2f b3 = *(const v2f*)(bR3 + k);
            acc0 = __builtin_amdgcn_wmma_f32_16x16x4_f32(false, a, false, b0, (short)0, acc0, false, false);
            acc1 = __builtin_amdgcn_wmma_f32_16x16x4_f32(false, a, false, b1, (short)0, acc1, false, false);
            acc2 = __builtin_amdgcn_wmma_f32_16x16x4_f32(false, a, false, b2, (short)0, acc2, false, false);
            acc3 = __builtin_amdgcn_wmma_f32_16x16x4_f32(false, a, false, b3, (short)0, acc3, false, false);
        }
    } else {
        const float* bCol = B + n0 + l16;     // B[k][n]: stride ldb in k
        #pragma unroll 2
        for (int k = k0; k < kend; k += 4) {
            v2f a = *(const v2f*)(aRow + k);
            const int r0 = (k + (hlf << 1)) * ldb;
            const int r1 = r0 + ldb;
            v2f b0, b1, b2, b3;
            b0.x = bCol[r0];      b0.y = bCol[r1];
            b1.x = bCol[r0 + 16]; b1.y = bCol[r1 + 16];
            b2.x = bCol[r0 + 32]; b2.y = bCol[r1 + 32];
            b3.x = bCol[r0 + 48]; b3.y = bCol[r1 + 48];
            acc0 = __builtin_amdgcn_wmma_f32_16x16x4_f32(false, a, false, b0, (short)0, acc0, false, false);
            acc1 = __builtin_amdgcn_wmma_f32_16x16x4_f32(false, a, false, b1, (short)0, acc1, false, false);
            acc2 = __builtin_amdgcn_wmma_f32_16x16x4_f32(false, a, false, b2, (short)0, acc2, false, false);
            acc3 = __builtin_amdgcn_wmma_f32_16x16x4_f32(false, a, false, b3, (short)0, acc3, false, false);
        }
    }

    // C/D layout: VGPR v -> row m0 + v + 8*hlf, col (n0 + 16j) + l16.
    const int rb = m0 + (hlf << 3);
    v8f accs[4] = {acc0, acc1, acc2, acc3};
    if (gridDim.y == 1) {
        #pragma unroll
        for (int j = 0; j < 4; ++j) {
            const int col = n0 + (j << 4) + l16;
            const float bval = bias ? bias[col] : 0.0f;
            #pragma unroll
            for (int v = 0; v < 8; ++v)
                D[(rb + v) * ldd + col] = accs[j][v] + bval;
        }
    } else {
        #pragma unroll
        for (int j = 0; j < 4; ++j) {
            const int col = n0 + (j << 4) + l16;
            #pragma unroll
            for (int v = 0; v < 8; ++v)
                atomicAdd(&D[(rb + v) * ldd + col], accs[j][v]);
        }
    }
}

// out[h*SEQ+s] = max(sqrt(max(x^T G x, 0)), 1e-8), x = X[s, h*64 : (h+1)*64]
__global__ __launch_bounds__(256) void qform_norm(
    const float* __restrict__ X, const float* __restrict__ G,
    float* __restrict__ out)
{
    __shared__ float g[HDIM * HDIM];
    for (int i = threadIdx.x; i < HDIM * HDIM; i += 256) g[i] = G[i];
    __syncthreads();

    const int gid = blockIdx.x * 256 + threadIdx.x;   // h*SEQ + s
    const int h = gid / SEQ;
    const int s = gid % SEQ;
    const float* x = X + s * HIDN + h * HDIM;

    float xv[HDIM];
    #pragma unroll
    for (int i = 0; i < HDIM; ++i) xv[i] = x[i];

    float acc = 0.0f;
    for (int i = 0; i < HDIM; ++i) {
        const float* gr = &g[i * HDIM];
        float gi = 0.0f;
        #pragma unroll
        for (int j = 0; j < HDIM; ++j) gi = fmaf(gr[j], xv[j], gi);
        acc = fmaf(xv[i], gi, acc);
    }
    out[gid] = fmaxf(sqrtf(fmaxf(acc, 0.0f)), 1e-8f);
}

// Cosine-normalize then softmax each (h,q) row of 512, in place.
__global__ __launch_bounds__(256) void softmax_rows(
    float* __restrict__ sim, const float* __restrict__ nq,
    const float* __restrict__ nk)
{
    const int row = blockIdx.x;          // h*SEQ + q
    const int h   = row / SEQ;
    float* p = sim + (size_t)row * SEQ;
    const float* nkh = nk + h * SEQ;
    const float invq = 1.0f / nq[row];

    __shared__ float red[256];
    const int t = threadIdx.x;

    float v0 = p[t]       * invq / nkh[t];
    float v1 = p[t + 256] * invq / nkh[t + 256];

    red[t] = fmaxf(v0, v1);
    __syncthreads();
    for (int s = 128; s > 0; s >>= 1) {
        if (t < s) red[t] = fmaxf(red[t], red[t + s]);
        __syncthreads();
    }
    const float mx = red[0];
    __syncthreads();

    const float e0 = __expf(v0 - mx);
    const float e1 = __expf(v1 - mx);
    red[t] = e0 + e1;
    __syncthreads();
    for (int s = 128; s > 0; s >>= 1) {
        if (t < s) red[t] += red[t + s];
        __syncthreads();
    }
    const float inv = 1.0f / red[0];
    p[t]       = e0 * inv;
    p[t + 256] = e1 * inv;
}

// out[i] = mean_h attn[h][i]
__global__ __launch_bounds__(256) void attn_mean(
    const float* __restrict__ attn, float* __restrict__ out)
{
    const int i = blockIdx.x * 256 + threadIdx.x;
    const int n = SEQ * SEQ;
    if (i < n) {
        float s = 0.0f;
        #pragma unroll
        for (int h = 0; h < NHEAD; ++h) s += attn[(size_t)h * n + i];
        out[i] = s * (1.0f / NHEAD);
    }
}

extern "C" void kernel_launch(void* const* d_in, const int* in_sizes, int n_in,
                              void* d_out, int out_size, void* d_ws, size_t ws_size,
                              hipStream_t stream)
{
    (void)in_sizes; (void)n_in; (void)out_size; (void)ws_size;

    const float* query  = (const float*)d_in[0];
    const float* key_   = (const float*)d_in[1];
    const float* value  = (const float*)d_in[2];
    const float* Wq = (const float*)d_in[3];
    const float* bq = (const float*)d_in[4];
    const float* Wk = (const float*)d_in[5];
    const float* bk = (const float*)d_in[6];
    const float* Wv = (const float*)d_in[7];
    const float* bv = (const float*)d_in[8];
    const float* q_base = (const float*)d_in[9];
    const float* k_base = (const float*)d_in[10];
    const float* v_base = (const float*)d_in[11];
    const float* Wh = (const float*)d_in[12];
    const float* bh = (const float*)d_in[13];
    const float* Wo = (const float*)d_in[14];
    const float* bo = (const float*)d_in[15];

    // Workspace layout (floats): ~3.70M floats = ~14.2 MB.
    float* ws  = (float*)d_ws;
    float* Qf  = ws;                 // [512,512]; head h = cols h*64..h*64+63
    float* Kf  = Qf  + SEQ * HIDN;
    float* Vf  = Kf  + SEQ * HIDN;
    float* Gkq = Vf  + SEQ * HIDN;   // [64,64] x4, contiguous for one memset
    float* Gqq = Gkq + HDIM * HDIM;
    float* Gkk = Gqq + HDIM * HDIM;
    float* Mt  = Gkk + HDIM * HDIM;
    float* QG  = Mt  + HDIM * HDIM;  // [512,512]
    float* VM  = QG  + SEQ * HIDN;   // [512,512]
    float* SIM = VM  + SEQ * HIDN;   // [8,512,512] scores -> attn (in place)
    float* nq  = SIM + (size_t)NHEAD * SEQ * SEQ;  // [8,512]
    float* nk  = nq  + NHEAD * SEQ;
    float* CTX = nk  + NHEAD * SEQ;  // [512,512]

    float* out  = (float*)d_out;       // [512,512]
    float* outm = out + SEQ * SEQ;     // attn.mean over heads [512,512]

    const dim3 b32(32);
    const int HS = SEQ * SEQ;

    // Zero Gram buffers (split-K accumulates via atomicAdd).
    hipMemsetAsync(Gkq, 0, 4 * HDIM * HDIM * sizeof(float), stream);

    // Tile grids: x = (M/16) * (N/64)
    const dim3 gProj(32 * 8, 1, 1);          // 512x512
    const dim3 gGram(4 * 1, 25, 1);          // 64x64, split-K 25x400
    const dim3 gHead(32 * 1, 1, NHEAD);      // 512x64 per head
    const dim3 gSim (32 * 8, 1, NHEAD);      // 512x512 per head

    // 1) QKV projections: X @ W^T + b
    gemm_wmma<true><<<gProj, b32, 0, stream>>>(
        query, HIDN, 0, Wq, HIDN, 0, Qf, HIDN, 0, bq, SEQ, HIDN, HIDN);
    gemm_wmma<true><<<gProj, b32, 0, stream>>>(
        key_,  HIDN, 0, Wk, HIDN, 0, Kf, HIDN, 0, bk, SEQ, HIDN, HIDN);
    gemm_wmma<true><<<gProj, b32, 0, stream>>>(
        value, HIDN, 0, Wv, HIDN, 0, Vf, HIDN, 0, bv, SEQ, HIDN, HIDN);

    // 2) Gram matrices over DHYP (split-K: 25 slices of 400).
    gemm_wmma<true><<<gGram, b32, 0, stream>>>(
        k_base, DHYP_, 0, q_base, DHYP_, 0, Gkq, HDIM, 0, nullptr, HDIM, HDIM, DHYP_);
    gemm_wmma<true><<<gGram, b32, 0, stream>>>(
        q_base, DHYP_, 0, q_base, DHYP_, 0, Gqq, HDIM, 0, nullptr, HDIM, HDIM, DHYP_);
    gemm_wmma<true><<<gGram, b32, 0, stream>>>(
        k_base, DHYP_, 0, k_base, DHYP_, 0, Gkk, HDIM, 0, nullptr, HDIM, HDIM, DHYP_);
    gemm_wmma<true><<<gGram, b32, 0, stream>>>(
        Wh, DHYP_, 0, v_base, DHYP_, 0, Mt, HDIM, 0, nullptr, HDIM, HDIM, DHYP_);

    // 3) QG_h = Q_h @ G_kq^T  (heads batched in grid.z, 64-col slices)
    gemm_wmma<true><<<gHead, b32, 0, stream>>>(
        Qf, HIDN, HDIM, Gkq, HDIM, 0, QG, HIDN, HDIM, nullptr, SEQ, HDIM, HDIM);

    // 4) Row norms |Qh|, |Kh|
    qform_norm<<<dim3(NHEAD * SEQ / 256), dim3(256), 0, stream>>>(Qf, Gqq, nq);
    qform_norm<<<dim3(NHEAD * SEQ / 256), dim3(256), 0, stream>>>(Kf, Gkk, nk);

    // 5) sim_h = QG_h @ K_h^T
    gemm_wmma<true><<<gSim, b32, 0, stream>>>(
        QG, HIDN, HDIM, Kf, HIDN, HDIM, SIM, SEQ, HS, nullptr, SEQ, SEQ, HDIM);

    // 6) Cosine normalize + softmax (in place)
    softmax_rows<<<dim3(NHEAD * SEQ), dim3(256), 0, stream>>>(SIM, nq, nk);

    // 7) VM_h = V_h @ Mt^T
    gemm_wmma<true><<<gHead, b32, 0, stream>>>(
        Vf, HIDN, HDIM, Mt, HDIM, 0, VM, HIDN, HDIM, nullptr, SEQ, HDIM, HDIM);

    // 8) ctx_h = attn_h @ VM_h + bh  (B non-transposed), into CTX col slices
    gemm_wmma<false><<<gHead, b32, 0, stream>>>(
        SIM, SEQ, HS, VM, HIDN, HDIM, CTX, HIDN, HDIM, bh, SEQ, HDIM, SEQ);

    // 9) out = CTX @ Wo^T + bo
    gemm_wmma<true><<<gProj, b32, 0, stream>>>(
        CTX, HIDN, 0, Wo, HIDN, 0, out, HIDN, 0, bo, SEQ, HIDN, HIDN);

    // 10) attention mean over heads
    attn_mean<<<dim3(HS / 256), dim3(256), 0, stream>>>(SIM, outm);
}